// SGBlock_26371099198321
// MI455X (gfx1250) — compile-verified
//
#include <hip/hip_runtime.h>
#include <math.h>

typedef float v2f __attribute__((ext_vector_type(2)));
typedef float v8f __attribute__((ext_vector_type(8)));

#define WMMAF32(a, b, c) \
  __builtin_amdgcn_wmma_f32_16x16x4_f32(false, (a), false, (b), (short)0, (c), false, false)

// LDS layout in floats
#define T_IN   0        // 10 tiles (8 x-channel + 2 z-packed), 16x16 each
#define TFT    2560     // fft-filtered tokens, token-major [u*8+ch]; later reused as RES
#define HID    5120     // 16x32 hidden staging (512)
#define TOK    5632     // LN2'd tokens, token-major [u*8+ch]
#define LDS_N  8192

__device__ __forceinline__ void ln8(float* t, const float* __restrict__ g,
                                    const float* __restrict__ b) {
  float m = 0.f;
#pragma unroll
  for (int i = 0; i < 8; ++i) m += t[i];
  m *= 0.125f;
  float v = 0.f;
#pragma unroll
  for (int i = 0; i < 8; ++i) { float d = t[i] - m; v += d * d; }
  v *= 0.125f;
  const float inv = 1.0f / sqrtf(v + 1e-5f);
#pragma unroll
  for (int i = 0; i < 8; ++i) t[i] = (t[i] - m) * inv * g[i] + b[i];
}

__device__ __forceinline__ float gelu_exact(float s) {
  return 0.5f * s * (1.0f + erff(s * 0.7071067811865476f));
}

// One 16x16 tile: Y = F T F (F = C - iS), Z = Y .* W(Hermitian-extended),
// X' = Re(conj(F) Z conj(F)).  C,S symmetric -> same chunk regs serve as A or B.
// Output scattered token-major into TFT.
__device__ __forceinline__ void fft_filter_tile(
    float* lds, int tin,
    const v2f* cC, const v2f* cS,
    const float* __restrict__ wfreq, int isZ, int chan, int lane)
{
  const int h = lane >> 4;
  const int n = lane & 15;
  const int SA = HID;          // scratch A (256)
  const int SB = HID + 256;    // scratch B (256)
  const v8f vz = {0.f,0.f,0.f,0.f,0.f,0.f,0.f,0.f};

  // ---- S1: U = T*C, V = T*S  (T as A-operand chunks from LDS)
  v8f U = vz, V = vz;
#pragma unroll
  for (int k = 0; k < 4; ++k) {
    const int col = 4 * k + 2 * h;
    v2f a; a.x = lds[tin + n * 16 + col]; a.y = lds[tin + n * 16 + col + 1];
    U = WMMAF32(a, cC[k], U);
    V = WMMAF32(a, cS[k], V);
  }
#pragma unroll
  for (int i = 0; i < 8; ++i) {
    lds[SA + (i + 8 * h) * 16 + n] = U[i];
    lds[SB + (i + 8 * h) * 16 + n] = V[i];
  }
  __syncthreads();

  // ---- S2: Yr = C*U - S*V ; R = C*V + S*U  (so Yi = -R)
  v8f P = vz, Q = vz, R = vz;
#pragma unroll
  for (int k = 0; k < 4; ++k) {
    const int row = 4 * k + 2 * h;
    v2f bU, bV;
    bU.x = lds[SA + row * 16 + n]; bU.y = lds[SA + (row + 1) * 16 + n];
    bV.x = lds[SB + row * 16 + n]; bV.y = lds[SB + (row + 1) * 16 + n];
    P = WMMAF32(cC[k], bU, P);
    Q = WMMAF32(cS[k], bV, Q);
    R = WMMAF32(cC[k], bV, R);
    R = WMMAF32(cS[k], bU, R);
  }
  __syncthreads();

  // ---- complex weight multiply; Hermitian mirror emulates irfft2's extension
#pragma unroll
  for (int i = 0; i < 8; ++i) {
    const int kw = i + 8 * h;
    const int kh = n;
    float wr, wi;
    if (!isZ) {
      if (kh <= 8) {
        const int idx = ((kw * 9 + kh) * 8 + chan) * 2;
        wr = wfreq[idx]; wi = wfreq[idx + 1];
      } else {
        const int idx = ((((16 - kw) & 15) * 9 + (16 - kh)) * 8 + chan) * 2;
        wr = wfreq[idx]; wi = -wfreq[idx + 1];
      }
    } else {
      const int cz = chan + 2 * (kw >> 3) + (kh >> 3);  // 2x2 channel blocks
      const int k8 = kw & 7, l8 = kh & 7;
      if (l8 <= 4) {
        const int idx = ((k8 * 5 + l8) * 8 + cz) * 2;
        wr = wfreq[idx]; wi = wfreq[idx + 1];
      } else {
        const int idx = ((((8 - k8) & 7) * 5 + (8 - l8)) * 8 + cz) * 2;
        wr = wfreq[idx]; wi = -wfreq[idx + 1];
      }
    }
    const float yr  = P[i] - Q[i];
    const float nyi = R[i];                      // -Yi
    lds[SA + (i + 8 * h) * 16 + n] = yr * wr + nyi * wi;   // Zr
    lds[SB + (i + 8 * h) * 16 + n] = yr * wi - nyi * wr;   // Zi
  }
  __syncthreads();

  // ---- S3: Pr = C*Zr - S*Zi ; Pi = C*Zi + S*Zr
  v8f A1 = vz, A2 = vz, A3 = vz;
#pragma unroll
  for (int k = 0; k < 4; ++k) {
    const int row = 4 * k + 2 * h;
    v2f bR, bI;
    bR.x = lds[SA + row * 16 + n]; bR.y = lds[SA + (row + 1) * 16 + n];
    bI.x = lds[SB + row * 16 + n]; bI.y = lds[SB + (row + 1) * 16 + n];
    A1 = WMMAF32(cC[k], bR, A1);
    A2 = WMMAF32(cS[k], bI, A2);
    A3 = WMMAF32(cC[k], bI, A3);
    A3 = WMMAF32(cS[k], bR, A3);
  }
  __syncthreads();
#pragma unroll
  for (int i = 0; i < 8; ++i) {
    lds[SA + (i + 8 * h) * 16 + n] = A1[i] - A2[i];  // Pr
    lds[SB + (i + 8 * h) * 16 + n] = A3[i];          // Pi
  }
  __syncthreads();

  // ---- S4: X' = Pr*C - Pi*S ; scatter token-major into TFT
  v8f B1 = vz, B2 = vz;
#pragma unroll
  for (int k = 0; k < 4; ++k) {
    const int col = 4 * k + 2 * h;
    v2f aR, aI;
    aR.x = lds[SA + n * 16 + col]; aR.y = lds[SA + n * 16 + col + 1];
    aI.x = lds[SB + n * 16 + col]; aI.y = lds[SB + n * 16 + col + 1];
    B1 = WMMAF32(aR, cC[k], B1);
    B2 = WMMAF32(aI, cS[k], B2);
  }
#pragma unroll
  for (int i = 0; i < 8; ++i) {
    const float val = B1[i] - B2[i];
    const int row = i + 8 * h;     // = w (x) / packed row (z)
    const int col = n;             // = h (x) / packed col (z)
    int u, ch;
    if (!isZ) { u = col * 16 + row; ch = chan; }
    else      { u = 256 + (col & 7) * 8 + (row & 7);
                ch = chan + 2 * (row >> 3) + (col >> 3); }
    lds[TFT + u * 8 + ch] = val;
  }
  __syncthreads();
}

__global__ void __launch_bounds__(32)
sgblock_kernel(const float* __restrict__ x,
               const float* __restrict__ n1g, const float* __restrict__ n1b,
               const float* __restrict__ n2g, const float* __restrict__ n2b,
               const float* __restrict__ n3g, const float* __restrict__ n3b,
               const float* __restrict__ n4g, const float* __restrict__ n4b,
               const float* __restrict__ c1w, const float* __restrict__ c1b,
               const float* __restrict__ c2w, const float* __restrict__ c2b,
               const float* __restrict__ m1w1, const float* __restrict__ m1b1,
               const float* __restrict__ m1w2, const float* __restrict__ m1b2,
               const float* __restrict__ m2w1, const float* __restrict__ m2b1,
               const float* __restrict__ m2w2, const float* __restrict__ m2b2,
               const float* __restrict__ cw,  const float* __restrict__ cwz,
               float* __restrict__ out)
{
  __shared__ float lds[LDS_N];
  const int lane = threadIdx.x;
  const int m   = lane & 15;          // row-in-tile / token-in-group / B-col
  const int h2  = (lane >> 4) * 2;    // 0 or 2 (K offset per half-wave)
  const int h8  = h2 << 2;            // 0 or 8 (C-layout row offset)
  const float* xb = x   + (size_t)blockIdx.x * 320 * 8;
  float*       ob = out + (size_t)blockIdx.x * 320 * 8;
  const v8f vz = {0.f,0.f,0.f,0.f,0.f,0.f,0.f,0.f};

  // ---- DFT constant operand chunks (A/B layouts coincide for symmetric matrices)
  v2f cCx[4], cSx[4], cCz[4], cSz[4];
#pragma unroll
  for (int k = 0; k < 4; ++k) {
#pragma unroll
    for (int v = 0; v < 2; ++v) {
      const int c = 4 * k + h2 + v;
      float s16, c16;
      sincosf(6.283185307179586f * (float)((m * c) & 15) * 0.0625f, &s16, &c16);
      ((float*)&cCx[k])[v] = 0.25f * c16;        // ortho: 1/sqrt(16) per dim
      ((float*)&cSx[k])[v] = 0.25f * s16;
      float vc = 0.f, vs = 0.f;
      if ((m >> 3) == (c >> 3)) {                // blockdiag(F8, F8)
        float s8, c8;
        sincosf(6.283185307179586f * (float)(((m & 7) * (c & 7)) & 7) * 0.125f, &s8, &c8);
        vc = 0.3535533905932738f * c8;           // 1/sqrt(8) per dim
        vs = 0.3535533905932738f * s8;
      }
      ((float*)&cCz[k])[v] = vc;
      ((float*)&cSz[k])[v] = vs;
    }
  }

  // ---- Phase 0: load tokens, LN1, scatter into tiles (tile[r=w][c=h])
  for (int it = 0; it < 8; ++it) {                       // x tokens 0..255
    const int nt = lane + it * 32;
    float t[8];
    const float4 lo = *(const float4*)(xb + (size_t)(64 + nt) * 8);
    const float4 hi = *(const float4*)(xb + (size_t)(64 + nt) * 8 + 4);
    t[0] = lo.x; t[1] = lo.y; t[2] = lo.z; t[3] = lo.w;
    t[4] = hi.x; t[5] = hi.y; t[6] = hi.z; t[7] = hi.w;
    ln8(t, n1g, n1b);
    const int r = nt & 15, c = nt >> 4;
#pragma unroll
    for (int ch = 0; ch < 8; ++ch)
      lds[T_IN + ch * 256 + r * 16 + c] = t[ch];
  }
  for (int it = 0; it < 2; ++it) {                       // z tokens 0..63
    const int nt = lane + it * 32;
    float t[8];
    const float4 lo = *(const float4*)(xb + (size_t)nt * 8);
    const float4 hi = *(const float4*)(xb + (size_t)nt * 8 + 4);
    t[0] = lo.x; t[1] = lo.y; t[2] = lo.z; t[3] = lo.w;
    t[4] = hi.x; t[5] = hi.y; t[6] = hi.z; t[7] = hi.w;
    ln8(t, n1g, n1b);
    const int hh = nt >> 3, ww = nt & 7;
#pragma unroll
    for (int ch = 0; ch < 8; ++ch) {
      const int row = ((ch >> 1) & 1) * 8 + ww;
      const int col = (ch & 1) * 8 + hh;
      lds[T_IN + (8 + (ch >> 2)) * 256 + row * 16 + col] = t[ch];
    }
  }
  __syncthreads();

  // ---- Phase 1: spectral filter (WMMA), writes TFT token-major
  for (int ch = 0; ch < 8; ++ch)
    fft_filter_tile(lds, T_IN + ch * 256, cCx, cSx, cw, 0, ch, lane);
  for (int zt = 0; zt < 2; ++zt)
    fft_filter_tile(lds, T_IN + (8 + zt) * 256, cCz, cSz, cwz, 1, 4 * zt, lane);

  // ---- Phase 2a: LN2 per token -> TOK (token-major)
  for (int it = 0; it < 10; ++it) {
    const int u = lane + it * 32;
    const float4 a = *(const float4*)&lds[TFT + u * 8];
    const float4 b = *(const float4*)&lds[TFT + u * 8 + 4];
    float t[8];
    t[0] = a.x; t[1] = a.y; t[2] = a.z; t[3] = a.w;
    t[4] = b.x; t[5] = b.y; t[6] = b.z; t[7] = b.w;
    ln8(t, n2g, n2b);
    float4 s0, s1;
    s0.x = t[0]; s0.y = t[1]; s0.z = t[2]; s0.w = t[3];
    s1.x = t[4]; s1.y = t[5]; s1.z = t[6]; s1.w = t[7];
    *(float4*)&lds[TOK + u * 8]     = s0;
    *(float4*)&lds[TOK + u * 8 + 4] = s1;
  }
  __syncthreads();

  // ---- Phase 2b (x branch): per 16-token group, MLP + conv3x3 fused on WMMA
  {
    // B operands in registers
    v2f B1t[2][2], B2t[8], TAP[9][2];
#pragma unroll
    for (int t = 0; t < 2; ++t)
#pragma unroll
      for (int k = 0; k < 2; ++k)
#pragma unroll
        for (int v = 0; v < 2; ++v)
          ((float*)&B1t[t][k])[v] = m1w1[(4 * k + h2 + v) * 32 + t * 16 + m];
#pragma unroll
    for (int k = 0; k < 8; ++k)
#pragma unroll
      for (int v = 0; v < 2; ++v)
        ((float*)&B2t[k])[v] = (m < 8) ? m1w2[(4 * k + h2 + v) * 8 + m] : 0.f;
#pragma unroll
    for (int t9 = 0; t9 < 9; ++t9)
#pragma unroll
      for (int k = 0; k < 2; ++k)
#pragma unroll
        for (int v = 0; v < 2; ++v)
          ((float*)&TAP[t9][k])[v] = (m < 8) ? c1w[(m * 8 + 4 * k + h2 + v) * 9 + t9] : 0.f;
    const float bn0 = m1b1[m];
    const float bn1 = m1b1[16 + m];

    for (int g = 0; g < 16; ++g) {
      // GEMM1: hidden = tok(16x8) @ w1(8x32)
      v8f H0 = vz, H1 = vz;
#pragma unroll
      for (int k = 0; k < 2; ++k) {
        const int base = TOK + (g * 16 + m) * 8 + 4 * k + h2;
        v2f a; a.x = lds[base]; a.y = lds[base + 1];
        H0 = WMMAF32(a, B1t[0][k], H0);
        H1 = WMMAF32(a, B1t[1][k], H1);
      }
#pragma unroll
      for (int i = 0; i < 8; ++i) {
        lds[HID + (i + h8) * 32 + m]      = gelu_exact(H0[i] + bn0);
        lds[HID + (i + h8) * 32 + 16 + m] = gelu_exact(H1[i] + bn1);
      }
      __syncthreads();
      // GEMM2 + conv accumulate into one C
      v8f acc = vz;
#pragma unroll
      for (int k = 0; k < 8; ++k) {
        const int base = HID + m * 32 + 4 * k + h2;
        v2f a; a.x = lds[base]; a.y = lds[base + 1];
        acc = WMMAF32(a, B2t[k], acc);
      }
#pragma unroll
      for (int dy = -1; dy <= 1; ++dy) {
        const int hs = g + dy;                   // uniform per group
        if (hs >= 0 && hs < 16) {
#pragma unroll
          for (int dx = -1; dx <= 1; ++dx) {
            const int t9 = (dy + 1) * 3 + (dx + 1);
            const int ws = m + dx;
            const bool ok = (ws >= 0) && (ws < 16);
#pragma unroll
            for (int k = 0; k < 2; ++k) {
              const int K0 = 4 * k + h2;
              v2f a;
              a.x = ok ? lds[T_IN + K0 * 256 + ws * 16 + hs] : 0.f;
              a.y = ok ? lds[T_IN + (K0 + 1) * 256 + ws * 16 + hs] : 0.f;
              acc = WMMAF32(a, TAP[t9][k], acc);
            }
          }
        }
      }
      if (m < 8) {
#pragma unroll
        for (int i = 0; i < 8; ++i)
          lds[TFT + (g * 16 + i + h8) * 8 + m] = acc[i];   // RES (reuses TFT)
      }
      __syncthreads();
    }
  }

  // ---- Phase 2b (z branch): 4 groups of 16 tokens (8x8 image, packed tiles)
  {
    v2f B1t[2][2], B2t[8], TAP[9][2];
#pragma unroll
    for (int t = 0; t < 2; ++t)
#pragma unroll
      for (int k = 0; k < 2; ++k)
#pragma unroll
        for (int v = 0; v < 2; ++v)
          ((float*)&B1t[t][k])[v] = m2w1[(4 * k + h2 + v) * 32 + t * 16 + m];
#pragma unroll
    for (int k = 0; k < 8; ++k)
#pragma unroll
      for (int v = 0; v < 2; ++v)
        ((float*)&B2t[k])[v] = (m < 8) ? m2w2[(4 * k + h2 + v) * 8 + m] : 0.f;
#pragma unroll
    for (int t9 = 0; t9 < 9; ++t9)
#pragma unroll
      for (int k = 0; k < 2; ++k)
#pragma unroll
        for (int v = 0; v < 2; ++v)
          ((float*)&TAP[t9][k])[v] = (m < 8) ? c2w[(m * 8 + 4 * k + h2 + v) * 9 + t9] : 0.f;
    const float bn0 = m2b1[m];
    const float bn1 = m2b1[16 + m];

    for (int zg = 0; zg < 4; ++zg) {
      v8f H0 = vz, H1 = vz;
#pragma unroll
      for (int k = 0; k < 2; ++k) {
        const int base = TOK + (256 + zg * 16 + m) * 8 + 4 * k + h2;
        v2f a; a.x = lds[base]; a.y = lds[base + 1];
        H0 = WMMAF32(a, B1t[0][k], H0);
        H1 = WMMAF32(a, B1t[1][k], H1);
      }
#pragma unroll
      for (int i = 0; i < 8; ++i) {
        lds[HID + (i + h8) * 32 + m]      = gelu_exact(H0[i] + bn0);
        lds[HID + (i + h8) * 32 + 16 + m] = gelu_exact(H1[i] + bn1);
      }
      __syncthreads();
      v8f acc = vz;
#pragma unroll
      for (int k = 0; k < 8; ++k) {
        const int base = HID + m * 32 + 4 * k + h2;
        v2f a; a.x = lds[base]; a.y = lds[base + 1];
        acc = WMMAF32(a, B2t[k], acc);
      }
      const int hh = 2 * zg + (m >> 3);
      const int ww = m & 7;
#pragma unroll
      for (int dy = -1; dy <= 1; ++dy) {
#pragma unroll
        for (int dx = -1; dx <= 1; ++dx) {
          const int t9 = (dy + 1) * 3 + (dx + 1);
          const int hs = hh + dy, ws = ww + dx;
          const bool ok = (hs >= 0) && (hs < 8) && (ws >= 0) && (ws < 8);
#pragma unroll
          for (int k = 0; k < 2; ++k) {
            v2f a;
#pragma unroll
            for (int v = 0; v < 2; ++v) {
              const int K = 4 * k + h2 + v;
              const int row = ((K >> 1) & 1) * 8 + ws;
              const int col = (K & 1) * 8 + hs;
              ((float*)&a)[v] = ok ? lds[T_IN + (8 + (K >> 2)) * 256 + row * 16 + col] : 0.f;
            }
            acc = WMMAF32(a, TAP[t9][k], acc);
          }
        }
      }
      if (m < 8) {
#pragma unroll
        for (int i = 0; i < 8; ++i)
          lds[TFT + (256 + zg * 16 + i + h8) * 8 + m] = acc[i];
      }
      __syncthreads();
    }
  }

  // ---- Phase 3: +biases, LN_out, relu, store
  for (int it = 0; it < 10; ++it) {
    const int u = lane + it * 32;
    const float4 a = *(const float4*)&lds[TFT + u * 8];
    const float4 b = *(const float4*)&lds[TFT + u * 8 + 4];
    float t[8];
    t[0] = a.x; t[1] = a.y; t[2] = a.z; t[3] = a.w;
    t[4] = b.x; t[5] = b.y; t[6] = b.z; t[7] = b.w;
    int ref;
    if (it < 8) {                                     // x branch
#pragma unroll
      for (int ch = 0; ch < 8; ++ch) t[ch] += m1b2[ch] + c1b[ch];
      ln8(t, n3g, n3b);
      ref = 64 + u;
    } else {                                          // z branch
#pragma unroll
      for (int ch = 0; ch < 8; ++ch) t[ch] += m2b2[ch] + c2b[ch];
      ln8(t, n4g, n4b);
      ref = u - 256;
    }
    float4 s0, s1;
    s0.x = fmaxf(t[0], 0.f); s0.y = fmaxf(t[1], 0.f);
    s0.z = fmaxf(t[2], 0.f); s0.w = fmaxf(t[3], 0.f);
    s1.x = fmaxf(t[4], 0.f); s1.y = fmaxf(t[5], 0.f);
    s1.z = fmaxf(t[6], 0.f); s1.w = fmaxf(t[7], 0.f);
    *(float4*)(ob + (size_t)ref * 8)     = s0;
    *(float4*)(ob + (size_t)ref * 8 + 4) = s1;
  }
}

extern "C" void kernel_launch(void* const* d_in, const int* in_sizes, int n_in,
                              void* d_out, int out_size, void* d_ws, size_t ws_size,
                              hipStream_t stream) {
  (void)n_in; (void)d_ws; (void)ws_size; (void)out_size;
  const int nB = in_sizes[0] / (320 * 8);   // 8192
  sgblock_kernel<<<nB, 32, 0, stream>>>(
      (const float*)d_in[0],
      (const float*)d_in[1],  (const float*)d_in[2],
      (const float*)d_in[3],  (const float*)d_in[4],
      (const float*)d_in[5],  (const float*)d_in[6],
      (const float*)d_in[7],  (const float*)d_in[8],
      (const float*)d_in[9],  (const float*)d_in[10],
      (const float*)d_in[11], (const float*)d_in[12],
      (const float*)d_in[13], (const float*)d_in[14],
      (const float*)d_in[15], (const float*)d_in[16],
      (const float*)d_in[17], (const float*)d_in[18],
      (const float*)d_in[19], (const float*)d_in[20],
      (const float*)d_in[21], (const float*)d_in[22],
      (float*)d_out);
}